// PairedSimplifiedGCN_2001454760607
// MI455X (gfx1250) — compile-verified
//
#include <hip/hip_runtime.h>

// PairedSimplifiedGCN collapsed to: edge-count build -> f32 WMMA GEMM
// P[128,256] = CNT[128,20000] @ x[20000,256] -> tiny WMMA chain with W0/W1/W2.

typedef __attribute__((ext_vector_type(2))) float v2f;
typedef __attribute__((ext_vector_type(8))) float v8f;

#define N_NODES_C  20000
#define N_EDGES_C  640000
#define IN_DIM_C   256
#define HID_C      128
#define N_GRAPHS_C 64
#define MROWS_C    128                       // 64 src rows + 64 dst rows
#define OUT_COLS_C 768                       // 3 layers * 2 * HID
#define KSPLIT_C   50
#define KCHUNK_C   (N_NODES_C / KSPLIT_C)    // 400, multiple of 4

static __device__ __forceinline__ v8f wmma_f32(v2f a, v2f b, v8f c) {
  // 8-arg form: (neg_a, A, neg_b, B, c_mod, C, reuse_a, reuse_b)
  return __builtin_amdgcn_wmma_f32_16x16x4_f32(false, a, false, b, (short)0, c,
                                               false, false);
}

// ---------------- workspace zeroing ----------------
__global__ void k_zero_i(int* __restrict__ p, int n) {
  int i = blockIdx.x * blockDim.x + threadIdx.x;
  if (i < n) p[i] = 0;
}

// ---------------- build CNT (node-major [20000][128]) ----------------
// cols 0..63  : src weights  cnt[n][g] += 1 for each edge with src=n (g=batch[n])
// cols 64..127: dst weights  cnt[n][64+g] += 1 for each edge with dst=n, g=batch[src]
__global__ void k_edges(const int* __restrict__ src, const int* __restrict__ dst,
                        const int* __restrict__ batch, int* __restrict__ cnti) {
  int e = blockIdx.x * blockDim.x + threadIdx.x;
  if (e >= N_EDGES_C) return;
  int s = src[e];
  int d = dst[e];
  int g = batch[s];
  atomicAdd(&cnti[s * MROWS_C + g], 1);
  atomicAdd(&cnti[d * MROWS_C + 64 + g], 1);
}

// ---------------- big GEMM: Ppart[z][m][f] = sum_{k in chunk z} CNT[k][m]*x[k][f]
// grid: (IN_DIM/32, MROWS/32, KSPLIT), block: 32 (one wave, EXEC all-ones)
__global__ void k_gemm_pool(const int* __restrict__ cnti, const float* __restrict__ x,
                            float* __restrict__ ppart) {
  const int lane = threadIdx.x;
  const int mm = lane & 15;        // A row / B col / C col within tile
  const int kh = lane >> 4;        // K-half select
  const int n0 = blockIdx.x * 32;
  const int m0 = blockIdx.y * 32;
  const int k0 = blockIdx.z * KCHUNK_C;

  v8f acc00 = {0.f,0.f,0.f,0.f,0.f,0.f,0.f,0.f};
  v8f acc01 = acc00, acc10 = acc00, acc11 = acc00;

  for (int k = k0; k < k0 + KCHUNK_C; k += 4) {
    const int kr0 = k + 2 * kh;
    const int kr1 = kr0 + 1;
    v2f a0, a1, b0, b1;
    // A[m][kk] = CNT[k+kk][m]  (CNT stored node-major [K][128])
    a0.x = (float)cnti[kr0 * MROWS_C + m0 + mm];
    a0.y = (float)cnti[kr1 * MROWS_C + m0 + mm];
    a1.x = (float)cnti[kr0 * MROWS_C + m0 + 16 + mm];
    a1.y = (float)cnti[kr1 * MROWS_C + m0 + 16 + mm];
    // B[kk][n] = x[k+kk][n]
    b0.x = x[kr0 * IN_DIM_C + n0 + mm];
    b0.y = x[kr1 * IN_DIM_C + n0 + mm];
    b1.x = x[kr0 * IN_DIM_C + n0 + 16 + mm];
    b1.y = x[kr1 * IN_DIM_C + n0 + 16 + mm];
    acc00 = wmma_f32(a0, b0, acc00);
    acc01 = wmma_f32(a0, b1, acc01);
    acc10 = wmma_f32(a1, b0, acc10);
    acc11 = wmma_f32(a1, b1, acc11);
  }

  float* outp = ppart + (size_t)blockIdx.z * (MROWS_C * IN_DIM_C);
  auto store_tile = [&](v8f acc, int mbase, int nbase) {
    union { v8f v; float f[8]; } u; u.v = acc;
#pragma unroll
    for (int v = 0; v < 8; ++v)
      outp[(mbase + v + 8 * kh) * IN_DIM_C + nbase + mm] = u.f[v];
  };
  store_tile(acc00, m0,      n0);
  store_tile(acc01, m0,      n0 + 16);
  store_tile(acc10, m0 + 16, n0);
  store_tile(acc11, m0 + 16, n0 + 16);
}

// ---------------- deterministic K-chunk reduction ----------------
__global__ void k_reduce_p(const float* __restrict__ ppart, float* __restrict__ P) {
  int i = blockIdx.x * blockDim.x + threadIdx.x;
  if (i >= MROWS_C * IN_DIM_C) return;
  float s = 0.f;
  for (int c = 0; c < KSPLIT_C; ++c) s += ppart[(size_t)c * (MROWS_C * IN_DIM_C) + i];
  P[i] = s;
}

// ---------------- per-graph edge counts + 1/max(c,1) ----------------
__global__ void k_counts(const int* __restrict__ cnti, float* __restrict__ cnts,
                         float* __restrict__ invc) {
  __shared__ int red[256];
  const int g = blockIdx.x;
  int s = 0;
  for (int n = threadIdx.x; n < N_NODES_C; n += 256) s += cnti[n * MROWS_C + g];
  red[threadIdx.x] = s;
  __syncthreads();
  for (int w = 128; w > 0; w >>= 1) {
    if (threadIdx.x < w) red[threadIdx.x] += red[threadIdx.x + w];
    __syncthreads();
  }
  if (threadIdx.x == 0) {
    float c = (float)red[0];
    cnts[g] = c;
    invc[g] = 1.0f / fmaxf(c, 1.0f);
  }
}

// ---------------- layer chain: Q = A@W + counts (x) bias; out slice = invc*Q
// A:[128][Kdim] row-major, W:[Kdim][128] row-major. grid (8,8), block 32.
__global__ void k_layer(const float* __restrict__ A, const float* __restrict__ W,
                        const float* __restrict__ bias, const float* __restrict__ cnts,
                        const float* __restrict__ invc, float* __restrict__ Q,
                        float* __restrict__ out, int Kdim, int layer) {
  const int lane = threadIdx.x;
  const int mm = lane & 15;
  const int kh = lane >> 4;
  const int n0 = blockIdx.x * 16;
  const int m0 = blockIdx.y * 16;

  v8f acc = {0.f,0.f,0.f,0.f,0.f,0.f,0.f,0.f};
  for (int k = 0; k < Kdim; k += 4) {
    const int kr0 = k + 2 * kh;
    v2f a, b;
    a.x = A[(m0 + mm) * Kdim + kr0];
    a.y = A[(m0 + mm) * Kdim + kr0 + 1];
    b.x = W[kr0 * HID_C + n0 + mm];
    b.y = W[(kr0 + 1) * HID_C + n0 + mm];
    acc = wmma_f32(a, b, acc);
  }

  union { v8f v; float f[8]; } u; u.v = acc;
#pragma unroll
  for (int v = 0; v < 8; ++v) {
    const int row = m0 + v + 8 * kh;          // 0..127
    const int col = n0 + mm;                  // 0..127
    const int g = row & 63;
    const int half = row >> 6;                // 0 = src block, 1 = dst block
    const float q = u.f[v] + cnts[g] * bias[col];
    Q[row * HID_C + col] = q;
    out[g * OUT_COLS_C + layer * 256 + half * HID_C + col] = invc[g] * q;
  }
}

extern "C" void kernel_launch(void* const* d_in, const int* in_sizes, int n_in,
                              void* d_out, int out_size, void* d_ws, size_t ws_size,
                              hipStream_t stream) {
  (void)in_sizes; (void)n_in; (void)out_size; (void)ws_size;
  const float* x     = (const float*)d_in[0];
  const int*   ei    = (const int*)d_in[1];
  const int*   batch = (const int*)d_in[2];
  const float* W0    = (const float*)d_in[3];
  const float* b0    = (const float*)d_in[4];
  const float* W1    = (const float*)d_in[5];
  const float* b1    = (const float*)d_in[6];
  const float* W2    = (const float*)d_in[7];
  const float* b2    = (const float*)d_in[8];
  const int* src = ei;
  const int* dst = ei + N_EDGES_C;
  float* out = (float*)d_out;

  // workspace layout (all 4-byte elements)
  const size_t CNTI_N  = (size_t)N_NODES_C * MROWS_C;            // 2,560,000
  const size_t PPART_N = (size_t)KSPLIT_C * MROWS_C * IN_DIM_C;  // 1,638,400
  const size_t P_N     = (size_t)MROWS_C * IN_DIM_C;             // 32,768
  const size_t Q_N     = (size_t)MROWS_C * HID_C;                // 16,384
  char* wsb = (char*)d_ws;
  int*   cnti  = (int*)wsb;                         wsb += CNTI_N  * 4;
  float* ppart = (float*)wsb;                       wsb += PPART_N * 4;
  float* P     = (float*)wsb;                       wsb += P_N     * 4;
  float* cnts  = (float*)wsb;                       wsb += 64 * 4;
  float* invc  = (float*)wsb;                       wsb += 64 * 4;
  float* Q0    = (float*)wsb;                       wsb += Q_N * 4;
  float* Q1    = (float*)wsb;                       wsb += Q_N * 4;
  float* Q2    = (float*)wsb;                       wsb += Q_N * 4;

  // 1) zero count matrix
  k_zero_i<<<(int)(CNTI_N / 256), 256, 0, stream>>>(cnti, (int)CNTI_N);
  // 2) per-edge integer atomics (exact, order-independent -> deterministic)
  k_edges<<<N_EDGES_C / 256, 256, 0, stream>>>(src, dst, batch, cnti);
  // 3) big f32 WMMA GEMM, K split into 50 chunks of 400
  k_gemm_pool<<<dim3(IN_DIM_C / 32, MROWS_C / 32, KSPLIT_C), 32, 0, stream>>>(cnti, x, ppart);
  // 4) fixed-order chunk reduction
  k_reduce_p<<<(int)(P_N / 256), 256, 0, stream>>>(ppart, P);
  // 5) per-graph counts + reciprocals
  k_counts<<<N_GRAPHS_C, 256, 0, stream>>>(cnti, cnts, invc);
  // 6-8) layer chain (each writes its 64x256 output slice)
  k_layer<<<dim3(8, 8), 32, 0, stream>>>(P,  W0, b0, cnts, invc, Q0, out, IN_DIM_C, 0);
  k_layer<<<dim3(8, 8), 32, 0, stream>>>(Q0, W1, b1, cnts, invc, Q1, out, HID_C, 1);
  k_layer<<<dim3(8, 8), 32, 0, stream>>>(Q1, W2, b2, cnts, invc, Q2, out, HID_C, 2);
}